// SupervisedHeteroGATModel_30631706755157
// MI455X (gfx1250) — compile-verified
//
#include <hip/hip_runtime.h>
#include <hip/hip_bf16.h>

#define NN 100000
#define FF 64
#define DD 128
#define HH 8
#define CC 16
#define EE 640000
#define OO 8

typedef __attribute__((ext_vector_type(2))) float v2f;
typedef __attribute__((ext_vector_type(8))) float v8f;

// ---------------------------------------------------------------- utilities
__global__ void fill_zero_kernel(float* __restrict__ p, long n) {
  long i = (long)blockIdx.x * blockDim.x + threadIdx.x;
  if (i < n) p[i] = 0.0f;
}

// ------------------------------------------------- WMMA f32 GEMM, M = 128
// C[nrows x 128] = A[nrows x K] @ W[K x 128] + bias[128]
// W is staged in LDS pre-swizzled into WMMA B-fragment order:
//   lds[((kstep*8 + t) << 6) + lane*2 + slot]
// so the inner-loop B reads are conflict-free ds_load_(2addr_)b64.
// One wave computes 32 rows x 128 cols (2 A fragments, 16 v8f accumulators);
// each B fragment feeds two WMMAs.
// __launch_bounds__(256, 1): ~170 VGPRs live (16 v8f accs + frags) — without
// this the compiler caps at 128 VGPRs and spills accumulators to scratch.
__global__ void __launch_bounds__(256, 1)
gemm128_wmma(const float* __restrict__ A, const float* __restrict__ W,
             const float* __restrict__ bias, float* __restrict__ C,
             int nrows, int K) {
  __shared__ float ldsW[DD * DD];       // up to 64 KB (K=128); 32 KB for K=64
  const int tid = threadIdx.x;

  // Stage W in fragment-order into LDS (coalesced global reads, one-time scatter).
  for (int i = tid; i < K * DD; i += 256) {
    int r = i >> 7;                     // row in W (k index)
    int c = i & 127;                    // col in W
    int kstep = r >> 2;
    int rr = r & 3;                     // 0..3 within the 16x16x4 step
    int lane = ((rr >> 1) << 4) + (c & 15);   // lanes 0-15: K={0,1}, 16-31: K={2,3}
    int t = c >> 4;                     // column tile
    int slot = rr & 1;                  // .x / .y of the v2f fragment
    ldsW[((kstep * 8 + t) << 6) + (lane << 1) + slot] = W[i];
  }
  __syncthreads();

  int wave = (int)(((long)blockIdx.x * blockDim.x + tid) >> 5);
  int lane = tid & 31;
  int row0 = wave << 5;                 // 32 rows per wave
  if (row0 >= nrows) return;            // wave-uniform; no further barriers

  v8f acc0[8], acc1[8];
  v8f vzero = {};
#pragma unroll
  for (int t = 0; t < 8; ++t) { acc0[t] = vzero; acc1[t] = vzero; }

  const int ll = lane & 15;
  const int kk = (lane >> 4) << 1;      // 0 for lanes 0-15, 2 for lanes 16-31
  const float* Ap0 = A + (size_t)(row0 + ll) * K + kk;
  const float* Ap1 = A + (size_t)(row0 + 16 + ll) * K + kk;
  const float* lb = ldsW + (lane << 1);

  v2f a0 = *(const v2f*)(Ap0);
  v2f a1 = *(const v2f*)(Ap1);

  const int ksteps = K >> 2;
  for (int ks = 0; ks < ksteps; ++ks) {
    // prefetch next A fragments (clamped on last iter; redundant but safe)
    const int kn = (ks + 1 < ksteps) ? (ks + 1) << 2 : ks << 2;
    v2f na0 = *(const v2f*)(Ap0 + kn);
    v2f na1 = *(const v2f*)(Ap1 + kn);

    v2f b[8];
#pragma unroll
    for (int t = 0; t < 8; ++t)
      b[t] = *(const v2f*)(lb + ((ks * 8 + t) << 6));
#pragma unroll
    for (int t = 0; t < 8; ++t) {
      acc0[t] = __builtin_amdgcn_wmma_f32_16x16x4_f32(
          false, a0, false, b[t], (short)0, acc0[t], false, false);
      acc1[t] = __builtin_amdgcn_wmma_f32_16x16x4_f32(
          false, a1, false, b[t], (short)0, acc1[t], false, false);
    }
    a0 = na0;
    a1 = na1;
  }

  // D layout: VGPR r -> row = row0 + r + (lane<16 ? 0 : 8), col = t*16 + (lane&15)
  const int rb = (lane >> 4) << 3;
#pragma unroll
  for (int t = 0; t < 8; ++t) {
    const int col = t * 16 + ll;
    const float bv = bias[col];
#pragma unroll
    for (int r = 0; r < 8; ++r) {
      C[(size_t)(row0 + rb + r) * DD + col] = acc0[t][r] + bv;
      C[(size_t)(row0 + 16 + rb + r) * DD + col] = acc1[t][r] + bv;
    }
  }
}

// ------------------------------------------------------------- edge phase
// Wave per edge: lane owns channels [4*lane, 4*lane+4), head = lane/4.
__global__ void edge_logits_kernel(const float* __restrict__ xl, const float* __restrict__ xr,
                                   const long long* __restrict__ ei, const float* __restrict__ eattr,
                                   const float* __restrict__ We, const float* __restrict__ att,
                                   float* __restrict__ logits, unsigned* __restrict__ nmax) {
  long gid = (long)blockIdx.x * blockDim.x + threadIdx.x;
  long e = gid >> 5;
  int lane = threadIdx.x & 31;
  if (e >= EE) return;
  long s = (long)ei[e];
  long d = (long)ei[EE + e];
  float ea = eattr[e];
  float4 a = ((const float4*)(xl + s * DD))[lane];
  float4 b = ((const float4*)(xr + d * DD))[lane];
  float4 w = ((const float4*)We)[lane];
  float4 t = ((const float4*)att)[lane];
  float e0 = a.x + b.x + ea * w.x;
  float e1 = a.y + b.y + ea * w.y;
  float e2 = a.z + b.z + ea * w.z;
  float e3 = a.w + b.w + ea * w.w;
  e0 = e0 > 0.f ? e0 : 0.2f * e0;
  e1 = e1 > 0.f ? e1 : 0.2f * e1;
  e2 = e2 > 0.f ? e2 : 0.2f * e2;
  e3 = e3 > 0.f ? e3 : 0.2f * e3;
  float part = e0 * t.x + e1 * t.y + e2 * t.z + e3 * t.w;
  part += __shfl_xor(part, 1, 32);
  part += __shfl_xor(part, 2, 32);
  if ((lane & 3) == 0) {
    int h = lane >> 2;
    logits[e * HH + h] = part;
    // order-preserving float->uint map; uint 0 == -inf sentinel
    unsigned u = __float_as_uint(part);
    unsigned m = (u & 0x80000000u) ? ~u : (u | 0x80000000u);
    atomicMax(nmax + d * HH + h, m);
  }
}

__global__ void edge_exp_kernel(const long long* __restrict__ ei, float* __restrict__ logits,
                                const unsigned* __restrict__ nmax, float* __restrict__ denom) {
  long i = (long)blockIdx.x * blockDim.x + threadIdx.x;
  if (i >= (long)EE * HH) return;
  long e = i >> 3;
  int h = (int)(i & 7);
  long d = (long)ei[EE + e];
  unsigned m = nmax[d * HH + h];
  float mx = __uint_as_float((m & 0x80000000u) ? (m & 0x7FFFFFFFu) : ~m);
  float ex = __expf(logits[i] - mx);
  logits[i] = ex;                       // in-place: numerator
  atomicAdd(denom + d * HH + h, ex);
}

__global__ void edge_scatter_kernel(const float* __restrict__ xl, const long long* __restrict__ ei,
                                    const float* __restrict__ exb, const float* __restrict__ denom,
                                    float* __restrict__ acc) {
  long gid = (long)blockIdx.x * blockDim.x + threadIdx.x;
  long e = gid >> 5;
  int lane = threadIdx.x & 31;
  if (e >= EE) return;
  long s = (long)ei[e];
  long d = (long)ei[EE + e];
  int h = lane >> 2;
  float alpha = exb[e * HH + h] / denom[d * HH + h];
  float4 v = ((const float4*)(xl + s * DD))[lane];
  float* ap = acc + d * DD + lane * 4;
  atomicAdd(ap + 0, alpha * v.x);
  atomicAdd(ap + 1, alpha * v.y);
  atomicAdd(ap + 2, alpha * v.z);
  atomicAdd(ap + 3, alpha * v.w);
}

__global__ void bias_act_kernel(const float* __restrict__ acc, const float* __restrict__ bias,
                                float* __restrict__ out, long n, int leaky) {
  long i = (long)blockIdx.x * blockDim.x + threadIdx.x;
  if (i >= n) return;
  float v = acc[i] + bias[i & (DD - 1)];
  if (leaky) v = v > 0.f ? v : 0.01f * v;
  out[i] = v;
}

// z = layernorm(acc + bias)*gamma + beta + x0   (wave per node)
__global__ void ln_residual_kernel(const float* __restrict__ acc, const float* __restrict__ bias,
                                   const float* __restrict__ gamma, const float* __restrict__ beta,
                                   const float* __restrict__ x0, float* __restrict__ z) {
  long gid = (long)blockIdx.x * blockDim.x + threadIdx.x;
  long n = gid >> 5;
  int lane = threadIdx.x & 31;
  if (n >= NN) return;
  float4 v = ((const float4*)(acc + n * DD))[lane];
  float4 bb = ((const float4*)bias)[lane];
  v.x += bb.x; v.y += bb.y; v.z += bb.z; v.w += bb.w;
  float s = v.x + v.y + v.z + v.w;
  float sq = v.x * v.x + v.y * v.y + v.z * v.z + v.w * v.w;
#pragma unroll
  for (int off = 1; off < 32; off <<= 1) {
    s += __shfl_xor(s, off, 32);
    sq += __shfl_xor(sq, off, 32);
  }
  float mean = s * (1.0f / DD);
  float var = sq * (1.0f / DD) - mean * mean;
  float rstd = rsqrtf(var + 1e-5f);
  float4 g = ((const float4*)gamma)[lane];
  float4 be = ((const float4*)beta)[lane];
  float4 x = ((const float4*)(x0 + n * DD))[lane];
  float4 o;
  o.x = (v.x - mean) * rstd * g.x + be.x + x.x;
  o.y = (v.y - mean) * rstd * g.y + be.y + x.y;
  o.z = (v.z - mean) * rstd * g.z + be.z + x.z;
  o.w = (v.w - mean) * rstd * g.w + be.w + x.w;
  ((float4*)(z + n * DD))[lane] = o;
}

// out[n,o] = z[n,:] @ W_out[:,o] + b_out[o]  (thread per (n,o); 0.2 GFLOP, negligible)
__global__ void head_kernel(const float* __restrict__ z, const float* __restrict__ Wo,
                            const float* __restrict__ bo, float* __restrict__ out) {
  long i = (long)blockIdx.x * blockDim.x + threadIdx.x;
  if (i >= (long)NN * OO) return;
  long n = i >> 3;
  int o = (int)(i & 7);
  const float* zp = z + n * DD;
  float sacc = bo[o];
#pragma unroll 8
  for (int k = 0; k < DD; ++k) sacc += zp[k] * Wo[k * OO + o];
  out[i] = sacc;
}

// ---------------------------------------------------------------- launcher
extern "C" void kernel_launch(void* const* d_in, const int* in_sizes, int n_in,
                              void* d_out, int out_size, void* d_ws, size_t ws_size,
                              hipStream_t stream) {
  const float* x        = (const float*)d_in[0];
  const long long* ei   = (const long long*)d_in[1];
  const float* eattr    = (const float*)d_in[2];
  const float* W_in     = (const float*)d_in[3];
  const float* b_in     = (const float*)d_in[4];
  const float* Wl1      = (const float*)d_in[5];
  const float* bl1      = (const float*)d_in[6];
  const float* Wr1      = (const float*)d_in[7];
  const float* br1      = (const float*)d_in[8];
  const float* We1      = (const float*)d_in[9];
  const float* att1     = (const float*)d_in[10];
  const float* bias1    = (const float*)d_in[11];
  const float* Wl2      = (const float*)d_in[12];
  const float* bl2      = (const float*)d_in[13];
  const float* Wr2      = (const float*)d_in[14];
  const float* br2      = (const float*)d_in[15];
  const float* We2      = (const float*)d_in[16];
  const float* att2     = (const float*)d_in[17];
  const float* bias2    = (const float*)d_in[18];
  const float* gamma    = (const float*)d_in[19];
  const float* beta     = (const float*)d_in[20];
  const float* W_out    = (const float*)d_in[21];
  const float* b_out    = (const float*)d_in[22];

  float* ws = (float*)d_ws;
  const size_t nd = (size_t)NN * DD;
  float*    x0     = ws;
  float*    hbuf   = ws + nd;              // layer-1 activations, reused for z
  float*    xl     = ws + 2 * nd;
  float*    xr     = ws + 3 * nd;
  float*    acc    = ws + 4 * nd;          // acc / nmax / denom contiguous
  unsigned* nmax   = (unsigned*)(ws + 5 * nd);
  float*    denom  = ws + 5 * nd + (size_t)NN * HH;
  float*    logits = ws + 5 * nd + 2 * (size_t)NN * HH;   // E*H floats

  const int TB = 256;
  const long fillcnt   = (long)nd + 2L * NN * HH;  // acc + nmax + denom
  const int  fill_bl   = (int)((fillcnt + TB - 1) / TB);
  const int  gemm_bl   = (NN / 32 + 7) / 8;        // 8 waves/block, wave -> 32 rows
  const int  edge_bl   = (EE + 7) / 8;             // 8 edges/block (wave per edge)
  const long ehcnt     = (long)EE * HH;
  const int  eh_bl     = (int)((ehcnt + TB - 1) / TB);
  const int  nd_bl     = (int)((nd + TB - 1) / TB);
  const int  node_bl   = (NN + 7) / 8;             // wave per node
  const int  head_bl   = (int)(((long)NN * OO + TB - 1) / TB);

  // input projection
  gemm128_wmma<<<gemm_bl, TB, 0, stream>>>(x, W_in, b_in, x0, NN, FF);

  // -------- GATv2 layer 1 --------
  fill_zero_kernel<<<fill_bl, TB, 0, stream>>>(acc, fillcnt);
  gemm128_wmma<<<gemm_bl, TB, 0, stream>>>(x0, Wl1, bl1, xl, NN, DD);
  gemm128_wmma<<<gemm_bl, TB, 0, stream>>>(x0, Wr1, br1, xr, NN, DD);
  edge_logits_kernel<<<edge_bl, TB, 0, stream>>>(xl, xr, ei, eattr, We1, att1, logits, nmax);
  edge_exp_kernel<<<eh_bl, TB, 0, stream>>>(ei, logits, nmax, denom);
  edge_scatter_kernel<<<edge_bl, TB, 0, stream>>>(xl, ei, logits, denom, acc);
  bias_act_kernel<<<nd_bl, TB, 0, stream>>>(acc, bias1, hbuf, (long)nd, 1);

  // -------- GATv2 layer 2 --------
  fill_zero_kernel<<<fill_bl, TB, 0, stream>>>(acc, fillcnt);
  gemm128_wmma<<<gemm_bl, TB, 0, stream>>>(hbuf, Wl2, bl2, xl, NN, DD);
  gemm128_wmma<<<gemm_bl, TB, 0, stream>>>(hbuf, Wr2, br2, xr, NN, DD);
  edge_logits_kernel<<<edge_bl, TB, 0, stream>>>(xl, xr, ei, eattr, We2, att2, logits, nmax);
  edge_exp_kernel<<<eh_bl, TB, 0, stream>>>(ei, logits, nmax, denom);
  edge_scatter_kernel<<<edge_bl, TB, 0, stream>>>(xl, ei, logits, denom, acc);

  // layernorm + residual, then output head
  ln_residual_kernel<<<node_bl, TB, 0, stream>>>(acc, bias2, gamma, beta, x0, hbuf);
  head_kernel<<<head_bl, TB, 0, stream>>>(hbuf, W_out, b_out, (float*)d_out);
}